// GraphQNetwork_78417512890500
// MI455X (gfx1250) — compile-verified
//
#include <hip/hip_runtime.h>
#include <hip/hip_bf16.h>
#include <math.h>

typedef float v2f __attribute__((ext_vector_type(2)));
typedef float v8f __attribute__((ext_vector_type(8)));

#define NEG_SLOPE 0.2f
#define EPS_GN 1e-5f

// ---------- helpers ----------

__device__ __forceinline__ int fmap(float f) {
    int i = __float_as_int(f);
    return i ^ ((i >> 31) & 0x7fffffff);   // monotonic float->signed-int map (involution)
}
__device__ __forceinline__ float funmap(int m) {
    return __int_as_float(m ^ ((m >> 31) & 0x7fffffff));
}

// packed B operand for WMMA: cols 0..5 = Wl, cols 8..13 = Wr, rest 0; rows >= Din are 0
__device__ __forceinline__ float wcoef(int k, int col, int Din,
                                       const float* __restrict__ Wl,
                                       const float* __restrict__ Wr) {
    if (k >= Din) return 0.f;
    if (col < 6) return Wl[k * 6 + col];
    if (col >= 8 && col < 14) return Wr[k * 6 + (col - 8)];
    return 0.f;
}

// ---------- fills ----------

__global__ void fill_f32(float* __restrict__ p, float v, int count) {
    int i = blockIdx.x * blockDim.x + threadIdx.x;
    if (i < count) p[i] = v;
}
__global__ void fill_i32(int* __restrict__ p, int v, int count) {
    int i = blockIdx.x * blockDim.x + threadIdx.x;
    if (i < count) p[i] = v;
}

// ---------- self-loop attr (mean of incoming edge_attr per target) ----------

__global__ void loop_accum_kernel(const int* __restrict__ ei, const float* __restrict__ eattr,
                                  float* __restrict__ deg, float* __restrict__ loopsum, int E) {
    int i = blockIdx.x * blockDim.x + threadIdx.x;
    if (i >= E) return;
    int d = ei[E + i];
    const float4 ea = *(const float4*)(eattr + (size_t)i * 4);
    atomicAdd(&deg[d], 1.0f);
    atomicAdd(&loopsum[(size_t)d * 4 + 0], ea.x);
    atomicAdd(&loopsum[(size_t)d * 4 + 1], ea.y);
    atomicAdd(&loopsum[(size_t)d * 4 + 2], ea.z);
    atomicAdd(&loopsum[(size_t)d * 4 + 3], ea.w);
}

__global__ void loop_div_kernel(const float* __restrict__ deg, float* __restrict__ loopat, int N) {
    int v = blockIdx.x * blockDim.x + threadIdx.x;
    if (v >= N) return;
    float inv = 1.0f / fmaxf(deg[v], 1.0f);
#pragma unroll
    for (int c = 0; c < 4; c++) loopat[(size_t)v * 4 + c] *= inv;
}

// ---------- node transforms via f32 WMMA (16x16x4), fully straight-line ----------
// All feature rows are stride-8 (layers 2/3 have zero-padded cols 6,7), so K is
// always 8: exactly 2 WMMAs with unconditional b64 A-loads. Packed B (Wl cols
// 0-5, Wr cols 8-13) staged transposed in LDS. D tile staged back through LDS
// (rows padded to 17 floats: conflict-free) so each lane writes one complete
// output row with unconditional b128+b64 stores.

__global__ void xform_kernel(const float* __restrict__ h, int Din,
                             const float* __restrict__ Wl, const float* __restrict__ bl,
                             const float* __restrict__ Wr, const float* __restrict__ br,
                             float* __restrict__ xl, float* __restrict__ xr, int N) {
    __shared__ float Bt[16][8];        // Bt[col][k]
    __shared__ float Ct[8][16][17];    // per-wave D tile [wave][row][col], padded
    {
        const int t = threadIdx.x;
        if (t < 128) {
            const int col = t >> 3, k = t & 7;
            Bt[col][k] = wcoef(k, col, Din, Wl, Wr);
        }
    }
    __syncthreads();

    const int gt = blockIdx.x * blockDim.x + threadIdx.x;
    const int wave = gt >> 5;
    const int wv = threadIdx.x >> 5;
    const int lane = threadIdx.x & 31;
    const int row0 = wave * 16;
    if (row0 >= N) return;  // wave-uniform; no barriers after this point

    const int m = lane & 15;              // A: row, B/C/D: column
    const int khalf = (lane >> 4) << 1;   // lanes 16-31 hold K+2,K+3 halves
    const size_t rb = (size_t)(row0 + m) << 3;

    v8f acc = {};
#pragma unroll
    for (int k0 = 0; k0 < 8; k0 += 4) {
        const v2f a  = *(const v2f*)(h + rb + k0 + khalf);     // 8B-aligned
        const v2f bv = *(const v2f*)(&Bt[m][k0 + khalf]);      // LDS, 8B-aligned
        acc = __builtin_amdgcn_wmma_f32_16x16x4_f32(false, a, false, bv,
                                                    (short)0, acc, false, false);
    }

    // stage D tile: lane holds rows {rowadd..rowadd+7} x col m
    const int rowadd = (lane >> 4) << 3;
#pragma unroll
    for (int j = 0; j < 8; j++)
        Ct[wv][j + rowadd][m] = acc[j];
    __builtin_amdgcn_wave_barrier();   // DS is in-order within a wave; pin schedule

    // lanes 0-15 write xl row m; lanes 16-31 write xr row m
    const int r = row0 + m;
    if (lane < 16) {
        if (r < N) {
            float4 o0; float2 o1;
            o0.x = Ct[wv][m][0] + bl[0];
            o0.y = Ct[wv][m][1] + bl[1];
            o0.z = Ct[wv][m][2] + bl[2];
            o0.w = Ct[wv][m][3] + bl[3];
            o1.x = Ct[wv][m][4] + bl[4];
            o1.y = Ct[wv][m][5] + bl[5];
            *(float4*)(xl + ((size_t)r << 3))     = o0;
            *(float2*)(xl + ((size_t)r << 3) + 4) = o1;
        }
    } else {
        if (r < N) {
            float4 o0; float2 o1;
            o0.x = Ct[wv][m][8]  + br[0];
            o0.y = Ct[wv][m][9]  + br[1];
            o0.z = Ct[wv][m][10] + br[2];
            o0.w = Ct[wv][m][11] + br[3];
            o1.x = Ct[wv][m][12] + br[4];
            o1.y = Ct[wv][m][13] + br[5];
            *(float4*)(xr + ((size_t)r << 3))     = o0;
            *(float2*)(xr + ((size_t)r << 3) + 4) = o1;
        }
    }
}

// ---------- edge scoring (shared by both passes) ----------

__device__ __forceinline__ void edge_score(int i, int E,
                                           const int* __restrict__ ei,
                                           const float* __restrict__ eattr,
                                           const float* __restrict__ loopat,
                                           const float* __restrict__ xl,
                                           const float* __restrict__ xr,
                                           const float* __restrict__ We,
                                           const float* __restrict__ att,
                                           int& s, int& d, float* xls, float* sc) {
    if (i < E) { s = ei[i]; d = ei[E + i]; }
    else       { s = d = i - E; }
    const float* eap = (i < E) ? (eattr + (size_t)i * 4) : (loopat + (size_t)s * 4);
    const float4 ea  = *(const float4*)eap;

    const float4 la  = *(const float4*)(xl + ((size_t)s << 3));
    const float2 lb  = *(const float2*)(xl + ((size_t)s << 3) + 4);
    const float4 ra  = *(const float4*)(xr + ((size_t)d << 3));
    const float2 rb2 = *(const float2*)(xr + ((size_t)d << 3) + 4);

    xls[0] = la.x; xls[1] = la.y; xls[2] = la.z; xls[3] = la.w; xls[4] = lb.x; xls[5] = lb.y;
    const float xrd[6] = { ra.x, ra.y, ra.z, ra.w, rb2.x, rb2.y };

#pragma unroll
    for (int hh = 0; hh < 2; hh++) {
        float acc = 0.f;
#pragma unroll
        for (int c = 0; c < 3; c++) {
            const int hc = hh * 3 + c;
            float mm = xls[hc] + xrd[hc] +
                       ea.x * We[hc] + ea.y * We[6 + hc] + ea.z * We[12 + hc] + ea.w * We[18 + hc];
            mm = (mm > 0.f) ? mm : NEG_SLOPE * mm;  // LeakyReLU(0.2)
            acc += mm * att[hc];
        }
        sc[hh] = acc;
    }
}

__global__ void edge_pass1_kernel(const int* __restrict__ ei, const float* __restrict__ eattr,
                                  const float* __restrict__ loopat,
                                  const float* __restrict__ xl, const float* __restrict__ xr,
                                  const float* __restrict__ We, const float* __restrict__ att,
                                  int* __restrict__ smax, int E, int EN) {
    int i = blockIdx.x * blockDim.x + threadIdx.x;
    if (i >= EN) return;
    int s, d;
    float xls[6], sc[2];
    edge_score(i, E, ei, eattr, loopat, xl, xr, We, att, s, d, xls, sc);
    atomicMax(&smax[(size_t)d * 2 + 0], fmap(sc[0]));
    atomicMax(&smax[(size_t)d * 2 + 1], fmap(sc[1]));
}

__global__ void edge_pass2_kernel(const int* __restrict__ ei, const float* __restrict__ eattr,
                                  const float* __restrict__ loopat,
                                  const float* __restrict__ xl, const float* __restrict__ xr,
                                  const float* __restrict__ We, const float* __restrict__ att,
                                  const int* __restrict__ smax,
                                  float* __restrict__ denom, float* __restrict__ attout,
                                  int E, int EN) {
    int i = blockIdx.x * blockDim.x + threadIdx.x;
    if (i >= EN) return;
    int s, d;
    float xls[6], sc[2];
    edge_score(i, E, ei, eattr, loopat, xl, xr, We, att, s, d, xls, sc);
#pragma unroll
    for (int hh = 0; hh < 2; hh++) {
        const float mx = funmap(smax[(size_t)d * 2 + hh]);
        const float ex = __expf(sc[hh] - mx);
        atomicAdd(&denom[(size_t)d * 2 + hh], ex);
#pragma unroll
        for (int c = 0; c < 3; c++)
            atomicAdd(&attout[(size_t)d * 6 + hh * 3 + c], ex * xls[hh * 3 + c]);
    }
}

__global__ void finalize_kernel(float* __restrict__ attout, const float* __restrict__ denom,
                                const float* __restrict__ bias, int N) {
    int v = blockIdx.x * blockDim.x + threadIdx.x;
    if (v >= N) return;
    const float d0 = 1.0f / denom[(size_t)v * 2 + 0];
    const float d1 = 1.0f / denom[(size_t)v * 2 + 1];
#pragma unroll
    for (int hc = 0; hc < 6; hc++)
        attout[(size_t)v * 6 + hc] = attout[(size_t)v * 6 + hc] * (hc < 3 ? d0 : d1) + bias[hc];
}

// ---------- GraphNorm + ReLU : one block per (contiguous) graph ----------
// Input stride 6 (attout), output stride 8 with zero-padded cols 6,7 so the
// next layer's WMMA A-operand padding is exactly zero.

__global__ void graphnorm_kernel(const float* __restrict__ g,
                                 const float* __restrict__ nw, const float* __restrict__ nb,
                                 const float* __restrict__ nms,
                                 float* __restrict__ out, int n) {
    const int b = blockIdx.x, tid = threadIdx.x, T = blockDim.x;
    const float* gb = g + (size_t)b * n * 6;
    float* ob = out + ((size_t)b * n << 3);
    __shared__ float red[256 * 6];
    __shared__ float mean[6], istd[6];

    float s[6] = {0, 0, 0, 0, 0, 0};
    for (int v = tid; v < n; v += T)
#pragma unroll
        for (int c = 0; c < 6; c++) s[c] += gb[(size_t)v * 6 + c];
#pragma unroll
    for (int c = 0; c < 6; c++) red[tid * 6 + c] = s[c];
    __syncthreads();
    for (int off = T >> 1; off > 0; off >>= 1) {
        if (tid < off)
#pragma unroll
            for (int c = 0; c < 6; c++) red[tid * 6 + c] += red[(tid + off) * 6 + c];
        __syncthreads();
    }
    if (tid < 6) mean[tid] = red[tid] / (float)n;
    __syncthreads();

#pragma unroll
    for (int c = 0; c < 6; c++) s[c] = 0.f;
    for (int v = tid; v < n; v += T)
#pragma unroll
        for (int c = 0; c < 6; c++) {
            const float o = gb[(size_t)v * 6 + c] - nms[c] * mean[c];
            s[c] += o * o;
        }
#pragma unroll
    for (int c = 0; c < 6; c++) red[tid * 6 + c] = s[c];
    __syncthreads();
    for (int off = T >> 1; off > 0; off >>= 1) {
        if (tid < off)
#pragma unroll
            for (int c = 0; c < 6; c++) red[tid * 6 + c] += red[(tid + off) * 6 + c];
        __syncthreads();
    }
    if (tid < 6) istd[tid] = rsqrtf(red[tid] / (float)n + EPS_GN);
    __syncthreads();

    for (int v = tid; v < n; v += T) {
#pragma unroll
        for (int c = 0; c < 6; c++) {
            const float o = gb[(size_t)v * 6 + c] - nms[c] * mean[c];
            float y = nw[c] * o * istd[c] + nb[c];
            ob[((size_t)v << 3) + c] = y > 0.f ? y : 0.f;  // fused ReLU
        }
        ob[((size_t)v << 3) + 6] = 0.f;   // zero pad for next layer's WMMA
        ob[((size_t)v << 3) + 7] = 0.f;
    }
}

// ---------- gather + dueling heads: single block, everything in LDS ----------

__global__ void heads_kernel(const float* __restrict__ x0, const float* __restrict__ f1,
                             const float* __restrict__ f2, const float* __restrict__ f3,
                             const int* __restrict__ cur, const int* __restrict__ amask, int n,
                             const float* __restrict__ vw1, const float* __restrict__ vb1,
                             const float* __restrict__ vw2, const float* __restrict__ vb2,
                             const float* __restrict__ vw3, const float* __restrict__ vb3,
                             const float* __restrict__ aw1, const float* __restrict__ ab1,
                             const float* __restrict__ aw2, const float* __restrict__ ab2,
                             const float* __restrict__ aw3, const float* __restrict__ ab3,
                             float* __restrict__ out) {
    const int tid = threadIdx.x;
    __shared__ float feat[64][26];
    __shared__ float a1[64][64];   // reused for adv3 output (cols < 50)
    __shared__ float a2[64][64];
    __shared__ float v1[64][16];
    __shared__ float v2s[64][8];
    __shared__ float v3s[64];
    __shared__ float madv[64];

    for (int idx = tid; idx < 64 * 26; idx += 256) {
        const int b = idx / 26, j = idx % 26;
        const int g = cur[b] + b * n;
        float val;
        if (j < 8)       val = x0[((size_t)g << 3) + j];
        else if (j < 14) val = f1[((size_t)g << 3) + (j - 8)];
        else if (j < 20) val = f2[((size_t)g << 3) + (j - 14)];
        else             val = f3[((size_t)g << 3) + (j - 20)];
        feat[b][j] = val;
    }
    __syncthreads();

    for (int idx = tid; idx < 64 * 64; idx += 256) {
        const int b = idx >> 6, o = idx & 63;
        float s = ab1[o];
        for (int k = 0; k < 26; k++) s += feat[b][k] * aw1[k * 64 + o];
        a1[b][o] = s > 0.f ? s : 0.f;
    }
    for (int idx = tid; idx < 64 * 16; idx += 256) {
        const int b = idx >> 4, o = idx & 15;
        float s = vb1[o];
        for (int k = 0; k < 26; k++) s += feat[b][k] * vw1[k * 16 + o];
        v1[b][o] = s > 0.f ? s : 0.f;
    }
    __syncthreads();

    for (int idx = tid; idx < 64 * 64; idx += 256) {
        const int b = idx >> 6, o = idx & 63;
        float s = ab2[o];
        for (int k = 0; k < 64; k++) s += a1[b][k] * aw2[k * 64 + o];
        a2[b][o] = s > 0.f ? s : 0.f;
    }
    for (int idx = tid; idx < 64 * 8; idx += 256) {
        const int b = idx >> 3, o = idx & 7;
        float s = vb2[o];
        for (int k = 0; k < 16; k++) s += v1[b][k] * vw2[k * 8 + o];
        v2s[b][o] = s > 0.f ? s : 0.f;
    }
    __syncthreads();

    for (int idx = tid; idx < 64 * 50; idx += 256) {
        const int b = idx / 50, o = idx % 50;
        float s = ab3[o];
        for (int k = 0; k < 64; k++) s += a2[b][k] * aw3[k * 50 + o];
        a1[b][o] = s;  // adv (no relu)
    }
    for (int idx = tid; idx < 64; idx += 256) {
        float s = vb3[0];
        for (int k = 0; k < 8; k++) s += v2s[idx][k] * vw3[k];
        v3s[idx] = s;
    }
    __syncthreads();

    if (tid < 64) {
        float m = 0.f;
        for (int o = 0; o < 50; o++) m += a1[tid][o];
        madv[tid] = m / 50.f;
    }
    __syncthreads();

    for (int idx = tid; idx < 64 * 50; idx += 256) {
        const int b = idx / 50, o = idx % 50;
        float q = v3s[b] + a1[b][o] - madv[b];
        if (amask[b * 50 + o] == 0) q = -1e8f;
        out[b * 50 + o] = q;
    }
}

// ---------- launcher ----------

extern "C" void kernel_launch(void* const* d_in, const int* in_sizes, int n_in,
                              void* d_out, int out_size, void* d_ws, size_t ws_size,
                              hipStream_t stream) {
    const float* x     = (const float*)d_in[0];
    const int*   ei    = (const int*)d_in[1];
    const float* eattr = (const float*)d_in[2];
    const int*   cur   = (const int*)d_in[4];
    const int*   amask = (const int*)d_in[5];

    const int N = in_sizes[0] / 8;
    const int E = in_sizes[2] / 4;
    const int B = in_sizes[4];
    const int n = N / B;

    // head params
    const float* vw1 = (const float*)d_in[36]; const float* vb1 = (const float*)d_in[37];
    const float* vw2 = (const float*)d_in[38]; const float* vb2 = (const float*)d_in[39];
    const float* vw3 = (const float*)d_in[40]; const float* vb3 = (const float*)d_in[41];
    const float* aw1 = (const float*)d_in[42]; const float* ab1 = (const float*)d_in[43];
    const float* aw2 = (const float*)d_in[44]; const float* ab2 = (const float*)d_in[45];
    const float* aw3 = (const float*)d_in[46]; const float* ab3 = (const float*)d_in[47];

    // workspace layout (floats), 55*N total
    float* ws     = (float*)d_ws;
    float* deg    = ws;                           // N
    float* loopat = deg + (size_t)N;              // 4N
    float* xl     = loopat + (size_t)4 * N;       // 8N (stride-8 rows, cols 0-5 used)
    float* xr     = xl + (size_t)8 * N;           // 8N
    int*   smax   = (int*)(xr + (size_t)8 * N);   // 2N (int)
    float* denom  = (float*)smax + (size_t)2 * N; // 2N
    float* attout = denom + (size_t)2 * N;        // 6N (stride-6)
    float* feats[3];
    feats[0] = attout + (size_t)6 * N;            // 8N each (stride-8, cols 6,7 = 0)
    feats[1] = feats[0] + (size_t)8 * N;
    feats[2] = feats[1] + (size_t)8 * N;

    const int TB = 256;
    auto blocks = [&](long long cnt) { return (unsigned)((cnt + TB - 1) / TB); };

    // self-loop edge_attr mean (shared across layers)
    fill_f32<<<blocks((long long)5 * N), TB, 0, stream>>>(deg, 0.f, 5 * N);
    loop_accum_kernel<<<blocks(E), TB, 0, stream>>>(ei, eattr, deg, loopat, E);
    loop_div_kernel<<<blocks(N), TB, 0, stream>>>(deg, loopat, N);

    const float* hin = x;   // N x 8, already stride-8
    int Din = 8;
    const int EN = E + N;
    const long long xform_threads = (long long)((N + 15) / 16) * 32;

    for (int l = 0; l < 3; l++) {
        const float* Wl  = (const float*)d_in[6 + 10 * l + 0];
        const float* bl  = (const float*)d_in[6 + 10 * l + 1];
        const float* Wr  = (const float*)d_in[6 + 10 * l + 2];
        const float* br  = (const float*)d_in[6 + 10 * l + 3];
        const float* We  = (const float*)d_in[6 + 10 * l + 4];
        const float* att = (const float*)d_in[6 + 10 * l + 5];
        const float* bb  = (const float*)d_in[6 + 10 * l + 6];
        const float* nw  = (const float*)d_in[6 + 10 * l + 7];
        const float* nb  = (const float*)d_in[6 + 10 * l + 8];
        const float* nm  = (const float*)d_in[6 + 10 * l + 9];

        xform_kernel<<<blocks(xform_threads), TB, 0, stream>>>(hin, Din,
                                                               Wl, bl, Wr, br, xl, xr, N);
        fill_i32<<<blocks((long long)2 * N), TB, 0, stream>>>(smax, (int)0x80000000, 2 * N);
        fill_f32<<<blocks((long long)8 * N), TB, 0, stream>>>(denom, 0.f, 8 * N);  // denom+attout

        edge_pass1_kernel<<<blocks(EN), TB, 0, stream>>>(ei, eattr, loopat, xl, xr,
                                                         We, att, smax, E, EN);
        edge_pass2_kernel<<<blocks(EN), TB, 0, stream>>>(ei, eattr, loopat, xl, xr,
                                                         We, att, smax, denom, attout, E, EN);
        finalize_kernel<<<blocks(N), TB, 0, stream>>>(attout, denom, bb, N);
        graphnorm_kernel<<<B, TB, 0, stream>>>(attout, nw, nb, nm, feats[l], n);

        hin = feats[l];
        Din = 6;
    }

    heads_kernel<<<1, 256, 0, stream>>>(x, feats[0], feats[1], feats[2], cur, amask, n,
                                        vw1, vb1, vw2, vb2, vw3, vb3,
                                        aw1, ab1, aw2, ab2, aw3, ab3,
                                        (float*)d_out);
}